// LearnableHighpass_67740224193135
// MI455X (gfx1250) — compile-verified
//
#include <hip/hip_runtime.h>

typedef float v2f __attribute__((ext_vector_type(2)));
typedef float v8f __attribute__((ext_vector_type(8)));

#ifndef HAVE_WMMA4
#ifdef __has_builtin
#if __has_builtin(__builtin_amdgcn_wmma_f32_16x16x4_f32)
#define HAVE_WMMA4 1
#endif
#endif
#endif
#ifndef HAVE_WMMA4
#define HAVE_WMMA4 0
#endif

#ifndef HAVE_ASYNC
#ifdef __has_builtin
#if __has_builtin(__builtin_amdgcn_global_load_async_to_lds_b128)
#define HAVE_ASYNC 1
#endif
#endif
#endif
#ifndef HAVE_ASYNC
#define HAVE_ASYNC 0
#endif

#if HAVE_ASYNC
#ifdef __has_builtin
#if __has_builtin(__builtin_amdgcn_s_wait_asynccnt)
#define ASYNC_WAIT(n) __builtin_amdgcn_s_wait_asynccnt(n)
#endif
#endif
#ifndef ASYNC_WAIT
#define ASYNC_WAIT(n) asm volatile("s_wait_asynccnt " #n ::: "memory")
#endif
// Builtin param type per hipcc diagnostic: GCC-vector int4 pointer.
typedef int v4i_b __attribute__((vector_size(4 * sizeof(int))));
typedef __attribute__((address_space(1))) v4i_b* gv4i_p;  // global (AS1)
typedef __attribute__((address_space(3))) v4i_b* lv4i_p;  // LDS (AS3)
// generic -> as(1): identity 64-bit address (inttoptr)
#define GAS(p) ((gv4i_p)(unsigned long long)(const void*)(p))
// generic -> as(3): low 32 bits of a flat LDS-aperture address are the LDS byte offset
#define LAS(p) ((lv4i_p)(unsigned)(unsigned long long)(const void*)(p))
#endif

#define SRATE   44100.0f
#define TLEN    131072
#define NROWS   128            // 64 batch * 2 channels
#define NSEG    64
#define SEGLEN  2048           // 64 * 2048 = 131072
#define TILE    256
#define NTILES  (SEGLEN / TILE)   // 8

__device__ __forceinline__ void hp_coefs(const float* __restrict__ ff,
                                         const float* __restrict__ fq,
                                         float& b0, float& b1, float& b2,
                                         float& a1, float& a2) {
  float f = fminf(fmaxf(ff[0], 100.0f), SRATE * 0.5f - 1.0f);
  float q = fminf(fmaxf(fq[0], 0.1f), 10.0f);
  float w0 = 2.0f * 3.14159265358979323846f * f / SRATE;
  float cw = cosf(w0);
  float alpha = sinf(w0) / (2.0f * q);
  float a0 = 1.0f + alpha;
  b0 = ((1.0f + cw) * 0.5f) / a0;
  b1 = (-(1.0f + cw)) / a0;
  b2 = b0;
  a1 = (-2.0f * cw) / a0;
  a2 = (1.0f - alpha) / a0;
}

// One wave (32 lanes) per segment. phase 0: compute zero-state end state of
// segment into ws_sum. phase 1: seed with ws_ent, write clamped output.
__global__ __launch_bounds__(32)
void hp_seg_kernel(const float* __restrict__ x,
                   const float* __restrict__ ff,
                   const float* __restrict__ fq,
                   float* __restrict__ out,
                   float* __restrict__ ws_sum,
                   const float* __restrict__ ws_ent,
                   int phase)
{
  // Double-buffered tile staging: X(i) = xb[4+i], i in [-2, 255]
  __shared__ float xbuf2[2][264] __attribute__((aligned(16)));
  __shared__ float ylT[256];   // ylT[n*16 + j] = zero-state resp, chunk j, sample n
  __shared__ float stb[32];    // per-chunk entry states (Y1,Y2)
  __shared__ float hh[16], pp[16], qq[16];

  const int lane = threadIdx.x;
  const int lo = lane & 15;    // A: row M base / B,D: column N
  const int hi = lane >> 4;    // lane-half selector
  const int blk = blockIdx.x;
  const int row = blk / NSEG;
  const int seg = blk % NSEG;
  const float* __restrict__ xrow = x + (size_t)row * TLEN;
  const int s0 = seg * SEGLEN;

  float b0, b1, b2, a1, a2;
  hp_coefs(ff, fq, b0, b1, b2, a1, a2);

#if HAVE_ASYNC
  // Pipeline prologue: async-stage tile 0 straight into LDS buffer 0
  {
    const float* xt = xrow + s0;
#pragma unroll
    for (int r = 0; r < 2; ++r) {
      int k = lane + 32 * r;
      __builtin_amdgcn_global_load_async_to_lds_b128(
          GAS(xt + 4 * k), LAS(&xbuf2[0][4 + 4 * k]), 0, 0);
    }
  }
#endif

  if (lane == 0) {
    // IIR impulse response h (zero-state kernel) and homogeneous solutions p,q
    hh[0] = 1.0f; hh[1] = -a1;
    for (int n = 2; n < 16; ++n) hh[n] = -a1 * hh[n - 1] - a2 * hh[n - 2];
    float pm1 = 1.0f, pm2 = 0.0f, qm1 = 0.0f, qm2 = 1.0f;
    for (int n = 0; n < 16; ++n) {
      float pn = -a1 * pm1 - a2 * pm2; pp[n] = pn; pm2 = pm1; pm1 = pn;
      float qn = -a1 * qm1 - a2 * qm2; qq[n] = qn; qm2 = qm1; qm1 = qn;
    }
  }
  __syncthreads();

  // A-matrix fragments: T[i][k] = (i>=k) ? h[i-k] : 0, 16x16 lower Toeplitz.
  // WMMA f32 16x16x4 A layout: lanes0-15 hold K=4kk+{0,1}, lanes16-31 K=4kk+{2,3}.
  v2f afrag[4];
#pragma unroll
  for (int kk = 0; kk < 4; ++kk) {
    int c = 4 * kk + 2 * hi;
    afrag[kk].x = (lo >= c)     ? hh[lo - c]     : 0.0f;
    afrag[kk].y = (lo >= c + 1) ? hh[lo - c - 1] : 0.0f;
  }
  const float p14 = pp[14], p15 = pp[15], q14 = qq[14], q15 = qq[15];

  float Y1 = 0.0f, Y2 = 0.0f;        // (y[-1], y[-2]) entering current chunk (lane0)
  if (phase == 1) { Y1 = ws_ent[2 * blk + 0]; Y2 = ws_ent[2 * blk + 1]; }

  // FIR carry (x[-1], x[-2]) across tile/segment boundary, read from global
  float cx1 = (s0 >= 1) ? xrow[s0 - 1] : 0.0f;
  float cx2 = (s0 >= 2) ? xrow[s0 - 2] : 0.0f;

  for (int tile = 0; tile < NTILES; ++tile) {
    float* xb = xbuf2[tile & 1];

#if HAVE_ASYNC
    if (tile + 1 < NTILES) {
      // Issue next tile into the other buffer, then wait for current tile
      // (the 2 outstanding async ops are the next tile's).
      const float* xn = xrow + s0 + (tile + 1) * TILE;
      float* xnb = xbuf2[(tile + 1) & 1];
#pragma unroll
      for (int r = 0; r < 2; ++r) {
        int k = lane + 32 * r;
        __builtin_amdgcn_global_load_async_to_lds_b128(
            GAS(xn + 4 * k), LAS(&xnb[4 + 4 * k]), 0, 0);
      }
      ASYNC_WAIT(2);
    } else {
      ASYNC_WAIT(0);
    }
#else
    const float4* xt4 = (const float4*)(xrow + s0 + tile * TILE);
#pragma unroll
    for (int r = 0; r < 2; ++r) {
      float4 v = xt4[lane + 32 * r];
      *(float4*)&xb[4 + 4 * (lane + 32 * r)] = v;
    }
    if (tile + 1 < NTILES) __builtin_prefetch(xt4 + 64, 0, 1);
#endif
    if (lane == 0) { xb[2] = cx2; xb[3] = cx1; }
    __syncthreads();
    cx2 = xb[4 + 254];
    cx1 = xb[4 + 255];

    v8f acc = {0, 0, 0, 0, 0, 0, 0, 0};
#if HAVE_WMMA4
#pragma unroll
    for (int kk = 0; kk < 4; ++kk) {
      // B fragment: B[k][j] = xf[j*16+k]; lanes0-15 K=4kk+{0,1}, lanes16-31 +2
      int base = lo * 16 + 4 * kk + 2 * hi;
      float xm2 = xb[2 + base];
      float xm1 = xb[3 + base];
      float xc0 = xb[4 + base];
      float xc1 = xb[5 + base];
      v2f bfrag;
      bfrag.x = b0 * xc0 + b1 * xm1 + b2 * xm2;
      bfrag.y = b0 * xc1 + b1 * xc0 + b2 * xm1;
      acc = __builtin_amdgcn_wmma_f32_16x16x4_f32(
          /*neg_a=*/false, afrag[kk], /*neg_b=*/false, bfrag,
          /*c_mod=*/(short)0, acc, /*reuse_a=*/false, /*reuse_b=*/false);
    }
#else
    // Layout-identical scalar fallback: acc[r] = D[r+8*hi][lo]
#pragma unroll
    for (int r = 0; r < 8; ++r) {
      int M = r + 8 * hi, N = lo;
      float sum = 0.0f;
      for (int k = 0; k <= M; ++k) {
        int i = N * 16 + k;
        float xfv = b0 * xb[4 + i] + b1 * xb[3 + i] + b2 * xb[2 + i];
        sum += hh[M - k] * xfv;
      }
      acc[r] = sum;
    }
#endif
    // D layout: VGPR r, lanes0-15 -> (M=r,N=lo); lanes16-31 -> (M=r+8,N=lo)
#pragma unroll
    for (int r = 0; r < 8; ++r) ylT[(r + 8 * hi) * 16 + lo] = acc[r];
    __syncthreads();

    // Serial 16-step chunk-state scan (lane 0)
    if (lane == 0) {
      float y1 = Y1, y2 = Y2;
#pragma unroll
      for (int j = 0; j < 16; ++j) {
        stb[2 * j + 0] = y1;
        stb[2 * j + 1] = y2;
        float yl15 = ylT[15 * 16 + j];
        float yl14 = ylT[14 * 16 + j];
        float n1 = yl15 + p15 * y1 + q15 * y2;
        float n2 = yl14 + p14 * y1 + q14 * y2;
        y1 = n1; y2 = n2;
      }
      Y1 = y1; Y2 = y2;
    }
    __syncthreads();

    if (phase == 1) {
      float* orow = out + (size_t)row * TLEN + s0 + tile * TILE;
#pragma unroll
      for (int r = 0; r < 8; ++r) {
        int idx = r * 32 + lane;
        int n = idx & 15, j = idx >> 4;
        float y = ylT[n * 16 + j] + pp[n] * stb[2 * j] + qq[n] * stb[2 * j + 1];
        y = fminf(fmaxf(y, -1.0f), 1.0f);
        orow[idx] = y;
      }
    }
  }

  if (phase == 0 && lane == 0) {
    ws_sum[2 * blk + 0] = Y1;   // y[SEGLEN-1] zero-state
    ws_sum[2 * blk + 1] = Y2;   // y[SEGLEN-2] zero-state
  }
}

// Inter-segment scan: one thread per row. s' = A^SEGLEN * s + v_seg.
__global__ __launch_bounds__(128)
void hp_scan_kernel(const float* __restrict__ ff,
                    const float* __restrict__ fq,
                    const float* __restrict__ ws_sum,
                    float* __restrict__ ws_ent)
{
  int row = threadIdx.x;
  if (row >= NROWS) return;
  float b0, b1, b2, a1, a2;
  hp_coefs(ff, fq, b0, b1, b2, a1, a2);
  // M = A^2048, A = [[-a1,-a2],[1,0]] via 11 squarings
  float m00 = -a1, m01 = -a2, m10 = 1.0f, m11 = 0.0f;
#pragma unroll
  for (int i = 0; i < 11; ++i) {
    float t00 = m00 * m00 + m01 * m10;
    float t01 = m00 * m01 + m01 * m11;
    float t10 = m10 * m00 + m11 * m10;
    float t11 = m10 * m01 + m11 * m11;
    m00 = t00; m01 = t01; m10 = t10; m11 = t11;
  }
  float s1 = 0.0f, s2 = 0.0f;
  for (int seg = 0; seg < NSEG; ++seg) {
    int i = row * NSEG + seg;
    ws_ent[2 * i + 0] = s1;
    ws_ent[2 * i + 1] = s2;
    float v1 = ws_sum[2 * i + 0];
    float v2 = ws_sum[2 * i + 1];
    float n1 = v1 + m00 * s1 + m01 * s2;
    float n2 = v2 + m10 * s1 + m11 * s2;
    s1 = n1; s2 = n2;
  }
}

extern "C" void kernel_launch(void* const* d_in, const int* in_sizes, int n_in,
                              void* d_out, int out_size, void* d_ws, size_t ws_size,
                              hipStream_t stream) {
  (void)in_sizes; (void)n_in; (void)out_size; (void)ws_size;
  const float* x  = (const float*)d_in[0];
  // d_in[1] is t (unused by reference)
  const float* ff = (const float*)d_in[2];
  const float* fq = (const float*)d_in[3];
  float* out = (float*)d_out;
  float* ws_sum = (float*)d_ws;                      // NROWS*NSEG*2 floats
  float* ws_ent = ws_sum + NROWS * NSEG * 2;         // NROWS*NSEG*2 floats

  dim3 grid(NROWS * NSEG), block(32);
  hp_seg_kernel<<<grid, block, 0, stream>>>(x, ff, fq, out, ws_sum, ws_ent, 0);
  hp_scan_kernel<<<1, 128, 0, stream>>>(ff, fq, ws_sum, ws_ent);
  hp_seg_kernel<<<grid, block, 0, stream>>>(x, ff, fq, out, ws_sum, ws_ent, 1);
}